// GST_GNN_Encoder_15650860827432
// MI455X (gfx1250) — compile-verified
//
#include <hip/hip_runtime.h>
#include <hip/hip_bf16.h>
#include <math.h>

// ---------------------------------------------------------------------------
// GST GNN encoder for MI455X (gfx1250, wave32).
//   Cost model: adj stream (2 x 256 MB @ 23.3 TB/s ~ 22us) dominates; the
//   dense GEMMs (~2.4 GFLOP) run on V_WMMA_F32_16X16X4_F32 (exact f32 math,
//   needed because the softmax exp() amplifies Wh error).
//   Attention exploits 0.4% sparsity: per-row nonzero list in LDS + online
//   masked softmax; adj read non-temporal so Wh/x1 stay resident in L2.
// ---------------------------------------------------------------------------

typedef float v2f __attribute__((ext_vector_type(2)));
typedef float v8f __attribute__((ext_vector_type(8)));

#define GAT_ALPHA 0.2f

// ---------------------------------------------------------------------------
// C(MxN) = A(MxK) @ B(KxN) [+ bias], f32 WMMA 16x16x4.
// TRANSB=1: B element [k][n] read from Bt[n*K + k] (for W_ih^T / W_hh^T).
// Block = 128 threads = 4 waves; wave w -> rows m0..m0+15;
// blockIdx.y -> 64-wide column tile (4 WMMA N-tiles per wave).
// K unrolled x2 (8 per iter): all fragment loads issued before the 8 WMMAs
// so global latency overlaps the matrix pipe. Requires M%64==0, N%64==0,
// K%8==0. EXEC all-1s throughout (no divergence before WMMA).
// ---------------------------------------------------------------------------
template <int TRANSB>
__global__ __launch_bounds__(128) void gemm_wmma_f32(
    const float* __restrict__ A, const float* __restrict__ B,
    const float* __restrict__ bias, float* __restrict__ C,
    int M, int N, int K)
{
    const int wave = threadIdx.x >> 5;
    const int lane = threadIdx.x & 31;
    const int m0   = blockIdx.x * 64 + wave * 16;
    const int n0   = blockIdx.y * 64;
    const int lr   = lane & 15;     // row-in-tile (A) / col-in-tile (B,C)
    const int hi   = lane >> 4;     // half-select
    const int koff = hi * 2;        // ISA A/B layout: lanes 16-31 hold K+2,K+3

    v8f acc[4] = {};
    const float* Arow = A + (size_t)(m0 + lr) * K + koff;

    for (int k = 0; k < K; k += 8) {
        // ---- issue all loads for two k-steps ----
        v2f a0 = *(const v2f*)(Arow + k);
        v2f a1 = *(const v2f*)(Arow + k + 4);
        v2f b0[4], b1[4];
#pragma unroll
        for (int t = 0; t < 4; ++t) {
            const int col = n0 + t * 16 + lr;
            if (TRANSB) {
                const float* Bc = B + (size_t)col * K + koff;
                b0[t] = *(const v2f*)(Bc + k);
                b1[t] = *(const v2f*)(Bc + k + 4);
            } else {
                const float* Bk = B + (size_t)(k + koff) * N + col;
                b0[t].x = Bk[0];
                b0[t].y = Bk[N];
                b1[t].x = Bk[4 * (size_t)N];
                b1[t].y = Bk[5 * (size_t)N];
            }
        }
        // ---- 8 WMMAs ----
#pragma unroll
        for (int t = 0; t < 4; ++t)
            acc[t] = __builtin_amdgcn_wmma_f32_16x16x4_f32(
                false, a0, false, b0[t], (short)0, acc[t], false, false);
#pragma unroll
        for (int t = 0; t < 4; ++t)
            acc[t] = __builtin_amdgcn_wmma_f32_16x16x4_f32(
                false, a1, false, b1[t], (short)0, acc[t], false, false);
    }

    // C/D layout: VGPR r -> M=r (lanes 0-15) / M=r+8 (lanes 16-31), N=lr.
#pragma unroll
    for (int t = 0; t < 4; ++t) {
        const int col = n0 + t * 16 + lr;
        const float bv = bias ? bias[col] : 0.0f;
#pragma unroll
        for (int r = 0; r < 8; ++r) {
            const int row = m0 + r + hi * 8;
            C[(size_t)row * N + col] = acc[t][r] + bv;
        }
    }
}

// ---------------------------------------------------------------------------
// src[i] = Wh[i,:] . a[0:128];  dst[i] = Wh[i,:] . a[128:256]
// One wave per row (wave32), 4 features per lane, shfl reduction.
// ---------------------------------------------------------------------------
__global__ __launch_bounds__(256) void attn_vec(
    const float* __restrict__ Wh, const float* __restrict__ a,
    float* __restrict__ src, float* __restrict__ dst)
{
    const int row  = blockIdx.x * 8 + (threadIdx.x >> 5);
    const int lane = threadIdx.x & 31;
    const float* w = Wh + (size_t)row * 128;
    float s = 0.0f, d = 0.0f;
#pragma unroll
    for (int q = 0; q < 4; ++q) {
        const int f = lane * 4 + q;
        const float v = w[f];
        s += v * a[f];
        d += v * a[128 + f];
    }
    for (int off = 16; off > 0; off >>= 1) {
        s += __shfl_down(s, off, 32);
        d += __shfl_down(d, off, 32);
    }
    if (lane == 0) { src[row] = s; dst[row] = d; }
}

// ---------------------------------------------------------------------------
// Fused masked softmax attention + ELU, one block (128 threads) per row.
//   out[i,f] = elu( sum_j softmax_j(lrelu(src_i+dst_j) | adj_ij>0) * Wh[j,f] )
// adj row streamed coalesced & non-temporal (512 MB total exceeds the 192 MB
// L2; streaming NT keeps the 4 MB Wh tables hot instead).
// ---------------------------------------------------------------------------
#define MAXNNZ 2048
__global__ __launch_bounds__(128) void gat_attention(
    const float* __restrict__ adj, const float* __restrict__ Wh,
    const float* __restrict__ src, const float* __restrict__ dst,
    float* __restrict__ out, int N)
{
    __shared__ int   s_j[MAXNNZ];
    __shared__ float s_e[MAXNNZ];
    __shared__ int   s_cnt;
    __shared__ float s_red[128];

    const int i = blockIdx.x;
    const int t = threadIdx.x;
    if (t == 0) s_cnt = 0;
    __syncthreads();

    const float  si   = src[i];
    const float* arow = adj + (size_t)i * N;

    // Pass 1: scan adjacency row (512B coalesced per iteration), collect
    // nonzeros with leaky-relu'd logits, track local max.
    float lmax = -3.0e38f;
    for (int j = t; j < N; j += 128) {
        __builtin_prefetch(arow + j + 1024, 0, 0);        // global_prefetch
        const float av = __builtin_nontemporal_load(arow + j);
        if (av > 0.0f) {
            float e = si + dst[j];
            e = (e > 0.0f) ? e : GAT_ALPHA * e;
            lmax = fmaxf(lmax, e);
            const int idx = atomicAdd(&s_cnt, 1);
            if (idx < MAXNNZ) { s_j[idx] = j; s_e[idx] = e; }
        }
    }

    // Block-reduce max (every row has >=1 nonzero: adj includes identity).
    s_red[t] = lmax;
    __syncthreads();
    for (int off = 64; off > 0; off >>= 1) {
        if (t < off) s_red[t] = fmaxf(s_red[t], s_red[t + off]);
        __syncthreads();
    }
    const float m = s_red[0];
    int nnz = s_cnt; nnz = (nnz < MAXNNZ) ? nnz : MAXNNZ;
    __syncthreads();   // everyone has read s_red[0] before reuse

    // Convert logits -> exp(e-m) in place; block-reduce the sum.
    float lsum = 0.0f;
    for (int idx = t; idx < nnz; idx += 128) {
        const float p = expf(s_e[idx] - m);
        s_e[idx] = p;
        lsum += p;
    }
    s_red[t] = lsum;
    __syncthreads();
    for (int off = 64; off > 0; off >>= 1) {
        if (t < off) s_red[t] += s_red[t + off];
        __syncthreads();
    }
    const float inv = 1.0f / s_red[0];

    // Pass 2: thread t owns feature f=t; Wh[j,*] reads are coalesced across
    // the block. Unroll x4 so several gather loads are in flight.
    float acc = 0.0f;
    int idx = 0;
    for (; idx + 4 <= nnz; idx += 4) {
        const float p0 = s_e[idx],     p1 = s_e[idx + 1];
        const float p2 = s_e[idx + 2], p3 = s_e[idx + 3];
        const float w0 = Wh[(size_t)s_j[idx]     * 128 + t];
        const float w1 = Wh[(size_t)s_j[idx + 1] * 128 + t];
        const float w2 = Wh[(size_t)s_j[idx + 2] * 128 + t];
        const float w3 = Wh[(size_t)s_j[idx + 3] * 128 + t];
        acc += p0 * w0 + p1 * w1 + p2 * w2 + p3 * w3;
    }
    for (; idx < nnz; ++idx)
        acc += s_e[idx] * Wh[(size_t)s_j[idx] * 128 + t];
    acc *= inv;
    out[(size_t)i * 128 + t] = (acc > 0.0f) ? acc : (expf(acc) - 1.0f);
}

// ---------------------------------------------------------------------------
// GRU gate combine: gi/gh are (N x 384) pre-activations (bias already added
// in the GEMM epilogue).
// ---------------------------------------------------------------------------
__global__ __launch_bounds__(256) void gru_combine(
    const float* __restrict__ gi, const float* __restrict__ gh,
    const float* __restrict__ h_prev, float* __restrict__ out)
{
    const int idx = blockIdx.x * 256 + threadIdx.x;
    const int i = idx >> 7, f = idx & 127;
    const size_t b = (size_t)i * 384;
    const float ir = gi[b + f], iz = gi[b + 128 + f], in_ = gi[b + 256 + f];
    const float hr = gh[b + f], hz = gh[b + 128 + f], hn  = gh[b + 256 + f];
    const float r = 1.0f / (1.0f + expf(-(ir + hr)));
    const float z = 1.0f / (1.0f + expf(-(iz + hz)));
    const float n = tanhf(in_ + r * hn);
    const float h = h_prev[(size_t)i * 128 + f];
    out[(size_t)i * 128 + f] = (1.0f - z) * n + z * h;
}

// ---------------------------------------------------------------------------
extern "C" void kernel_launch(void* const* d_in, const int* in_sizes, int n_in,
                              void* d_out, int out_size, void* d_ws, size_t ws_size,
                              hipStream_t stream) {
    const float* x      = (const float*)d_in[0];
    const float* adj    = (const float*)d_in[1];
    const float* h_prev = (const float*)d_in[2];
    const float* W1     = (const float*)d_in[3];
    const float* a1     = (const float*)d_in[4];
    const float* W2     = (const float*)d_in[5];
    const float* a2     = (const float*)d_in[6];
    const float* W_ih   = (const float*)d_in[7];
    const float* W_hh   = (const float*)d_in[8];
    const float* b_ih   = (const float*)d_in[9];
    const float* b_hh   = (const float*)d_in[10];
    float* out = (float*)d_out;
    float* ws  = (float*)d_ws;

    const int N = 8192, F0 = 256, H = 128, G = 384;
    const size_t M1 = (size_t)N * H;              // 1M floats
    float* Wh1  = ws;                             // N x H
    float* x1   = ws + 1 * M1;                    // N x H
    float* Wh2  = ws + 2 * M1;                    // N x H
    float* hsp  = ws + 3 * M1;                    // N x H
    float* gi   = ws + 4 * M1;                    // N x 384 (3M floats)
    float* gh   = ws + 7 * M1;                    // N x 384 (3M floats)
    float* src1 = ws + 10 * M1;
    float* dst1 = src1 + N;
    float* src2 = dst1 + N;
    float* dst2 = src2 + N;

    const dim3 blk(128);

    // ---- GAT layer 1 ----
    gemm_wmma_f32<0><<<dim3(N / 64, H / 64), blk, 0, stream>>>(x,  W1, nullptr, Wh1, N, H, F0);
    attn_vec        <<<N / 8, 256, 0, stream>>>(Wh1, a1, src1, dst1);
    gat_attention   <<<N, 128, 0, stream>>>(adj, Wh1, src1, dst1, x1, N);

    // ---- GAT layer 2 ----
    gemm_wmma_f32<0><<<dim3(N / 64, H / 64), blk, 0, stream>>>(x1, W2, nullptr, Wh2, N, H, H);
    attn_vec        <<<N / 8, 256, 0, stream>>>(Wh2, a2, src2, dst2);
    gat_attention   <<<N, 128, 0, stream>>>(adj, Wh2, src2, dst2, hsp, N);

    // ---- GRU cell ----
    gemm_wmma_f32<1><<<dim3(N / 64, G / 64), blk, 0, stream>>>(hsp,    W_ih, b_ih, gi, N, G, H);
    gemm_wmma_f32<1><<<dim3(N / 64, G / 64), blk, 0, stream>>>(h_prev, W_hh, b_hh, gh, N, G, H);
    gru_combine     <<<(N * H) / 256, 256, 0, stream>>>(gi, gh, h_prev, out);
}